// GCN_26499948216402
// MI455X (gfx1250) — compile-verified
//
#include <hip/hip_runtime.h>

// ---------------------------------------------------------------------------
// GCN on MI455X (gfx1250). Problem sizes are compile-time constants from the
// reference: N=100000 nodes, E=1600000 edges, D_IN=D_HID=64, D_OUT=16.
//
// Pipeline (all on `stream`):
//   deg  : float atomics over dst               -> dinv[i] = rsqrt(deg+1)
//   L1-3 : H = X @ Wl          (WMMA f32 16x16x4, fp32 exact)
//          AGG = 0; AGG[dst] += H[src]*dinv[src]*dinv[dst]   (f32 atomics)
//          AGG = relu(AGG + H*dinv^2 + b)       (fused epilogue, in place)
//   FFN  : H = relu(X @ Wf1 + bf1)  ;  OUT = H @ Wf2 + bf2   (WMMA GEMMs)
//
// Workspace layout (floats): [0, N)            dinv (also deg scratch)
//                            [N, N+N*64)       bufA
//                            [N+N*64, ...)     bufB     => ~51.6 MB needed.
// ---------------------------------------------------------------------------

typedef float v2f __attribute__((ext_vector_type(2)));
typedef float v8f __attribute__((ext_vector_type(8)));

#define NN   100000
#define EE   1600000
#define DH   64
#define DOUT 16

// ---------------------------------------------------------------- utilities
__global__ void gcn_zero_f32(float* __restrict__ p, int n) {
  int i = blockIdx.x * blockDim.x + threadIdx.x;
  if (i < n) p[i] = 0.0f;
}

__global__ void gcn_deg_kernel(const int* __restrict__ dst,
                               float* __restrict__ deg) {
  int e = blockIdx.x * blockDim.x + threadIdx.x;
  if (e < EE) unsafeAtomicAdd(&deg[dst[e]], 1.0f);
}

__global__ void gcn_dinv_kernel(float* __restrict__ deg) {
  int i = blockIdx.x * blockDim.x + threadIdx.x;
  if (i < NN) deg[i] = rsqrtf(deg[i] + 1.0f);
}

// ------------------------------------------------- WMMA fp32 GEMM (M x K=64)
// One wave (32 lanes) computes one 16x16 output tile of Out = X @ W (+bias,relu).
// X: [M x 64] row-major, W: [64 x NCOLS] row-major.
// A frag (16x4 f32): lanes 0-15 hold row M=l, VGPR0=K0 VGPR1=K1;
//                    lanes 16-31 hold row M=l, VGPR0=K2 VGPR1=K3.
// B frag (4x16 f32): mirrored with N in place of M.
// C/D (16x16 f32):   VGPR r holds row (r + 8*halfwave), col = lane&15.
template <int NCOLS, bool BIAS, bool RELU>
__global__ __launch_bounds__(32) void gcn_gemm_wmma(
    const float* __restrict__ X, const float* __restrict__ W,
    const float* __restrict__ bias, float* __restrict__ Out) {
  const int tileM = blockIdx.x;
  const int tileN = blockIdx.y;
  const int lane  = threadIdx.x;   // 0..31, wave32
  const int half  = lane >> 4;     // 0: lanes 0-15, 1: lanes 16-31
  const int l     = lane & 15;
  const int arow  = tileM * 16 + l;
  const int bcol  = tileN * 16 + l;

  v8f c = {};
#pragma unroll
  for (int k0 = 0; k0 < DH; k0 += 4) {
    const int ak = k0 + half * 2;
    v2f a, b;
    a.x = X[arow * DH + ak];
    a.y = X[arow * DH + ak + 1];
    b.x = W[ak * NCOLS + bcol];
    b.y = W[(ak + 1) * NCOLS + bcol];
    // D = A*B + C   (v_wmma_f32_16x16x4_f32)
    c = __builtin_amdgcn_wmma_f32_16x16x4_f32(
        /*neg_a=*/false, a, /*neg_b=*/false, b,
        /*c_mod=*/(short)0, c, /*reuse_a=*/false, /*reuse_b=*/false);
  }

#pragma unroll
  for (int r = 0; r < 8; ++r) {
    const int row = tileM * 16 + half * 8 + r;
    float v = c[r];
    if (BIAS) v += bias[bcol];
    if (RELU) v = fmaxf(v, 0.0f);
    Out[row * NCOLS + bcol] = v;
  }
}

// -------------------------------------------- edge scatter (normalized msgs)
// One thread per (edge, feature). Wave lanes cover 32 consecutive features of
// one edge -> coalesced 128B gather of H[src] and coalesced f32 atomics.
__global__ void gcn_scatter_kernel(const float* __restrict__ H,
                                   const int* __restrict__ src,
                                   const int* __restrict__ dst,
                                   const float* __restrict__ dinv,
                                   float* __restrict__ AGG) {
  int idx = blockIdx.x * blockDim.x + threadIdx.x;  // < EE*64 = 102.4M
  if (idx >= EE * DH) return;
  const int e = idx >> 6;
  const int d = idx & (DH - 1);
  const int s = src[e];
  const int t = dst[e];
  const float coef = dinv[s] * dinv[t];
  unsafeAtomicAdd(&AGG[t * DH + d], H[s * DH + d] * coef);
}

// ------------------------- self-loop + bias + relu epilogue (AGG in place)
__global__ void gcn_epilogue_kernel(float* __restrict__ AGG,
                                    const float* __restrict__ H,
                                    const float* __restrict__ dinv,
                                    const float* __restrict__ b) {
  int idx = blockIdx.x * blockDim.x + threadIdx.x;
  if (idx >= NN * DH) return;
  const int i = idx >> 6;
  const int d = idx & (DH - 1);
  const float di = dinv[i];
  const float v = AGG[idx] + H[idx] * di * di + b[d];
  AGG[idx] = fmaxf(v, 0.0f);
}

// ---------------------------------------------------------------------------
extern "C" void kernel_launch(void* const* d_in, const int* in_sizes, int n_in,
                              void* d_out, int out_size, void* d_ws,
                              size_t ws_size, hipStream_t stream) {
  (void)in_sizes; (void)n_in; (void)out_size; (void)ws_size;

  const float* x   = (const float*)d_in[0];
  const int*   ei  = (const int*)d_in[1];
  const int*   src = ei;        // edge_index[0]
  const int*   dst = ei + EE;   // edge_index[1]
  const float* W1  = (const float*)d_in[2];
  const float* b1  = (const float*)d_in[3];
  const float* W2  = (const float*)d_in[4];
  const float* b2  = (const float*)d_in[5];
  const float* W3  = (const float*)d_in[6];
  const float* b3  = (const float*)d_in[7];
  const float* Wf1 = (const float*)d_in[8];
  const float* bf1 = (const float*)d_in[9];
  const float* Wf2 = (const float*)d_in[10];
  const float* bf2 = (const float*)d_in[11];
  float* out = (float*)d_out;

  float* base = (float*)d_ws;
  float* dinv = base;             // NN floats (deg scratch, then dinv)
  float* bufA = base + NN;        // NN*64 floats
  float* bufB = bufA + NN * DH;   // NN*64 floats

  const int T = 256;
  const int gN   = (NN + T - 1) / T;
  const int gE   = (EE + T - 1) / T;
  const int gND  = (NN * DH + T - 1) / T;
  const int gED  = (EE * DH + T - 1) / T;
  const dim3 gemmGrid(NN / 16, DH / 16);     // 6250 x 4
  const dim3 gemmGridOut(NN / 16, DOUT / 16);  // 6250 x 1
  const dim3 gemmBlk(32);

  // degrees -> dinv
  gcn_zero_f32<<<gN, T, 0, stream>>>(dinv, NN);
  gcn_deg_kernel<<<gE, T, 0, stream>>>(dst, dinv);
  gcn_dinv_kernel<<<gN, T, 0, stream>>>(dinv);

  // ---- layer 1: in = x -> out in bufB
  gcn_gemm_wmma<DH, false, false><<<gemmGrid, gemmBlk, 0, stream>>>(x, W1, nullptr, bufA);
  gcn_zero_f32<<<gND, T, 0, stream>>>(bufB, NN * DH);
  gcn_scatter_kernel<<<gED, T, 0, stream>>>(bufA, src, dst, dinv, bufB);
  gcn_epilogue_kernel<<<gND, T, 0, stream>>>(bufB, bufA, dinv, b1);

  // ---- layer 2: in = bufB -> out in bufB (bufB dead after GEMM reads it)
  gcn_gemm_wmma<DH, false, false><<<gemmGrid, gemmBlk, 0, stream>>>(bufB, W2, nullptr, bufA);
  gcn_zero_f32<<<gND, T, 0, stream>>>(bufB, NN * DH);
  gcn_scatter_kernel<<<gED, T, 0, stream>>>(bufA, src, dst, dinv, bufB);
  gcn_epilogue_kernel<<<gND, T, 0, stream>>>(bufB, bufA, dinv, b2);

  // ---- layer 3
  gcn_gemm_wmma<DH, false, false><<<gemmGrid, gemmBlk, 0, stream>>>(bufB, W3, nullptr, bufA);
  gcn_zero_f32<<<gND, T, 0, stream>>>(bufB, NN * DH);
  gcn_scatter_kernel<<<gED, T, 0, stream>>>(bufA, src, dst, dinv, bufB);
  gcn_epilogue_kernel<<<gND, T, 0, stream>>>(bufB, bufA, dinv, b3);

  // ---- FFN: relu(bufB @ Wf1 + bf1) -> bufA ; bufA @ Wf2 + bf2 -> out
  gcn_gemm_wmma<DH, true, true><<<gemmGrid, gemmBlk, 0, stream>>>(bufB, Wf1, bf1, bufA);
  gcn_gemm_wmma<DOUT, true, false><<<gemmGridOut, gemmBlk, 0, stream>>>(bufA, Wf2, bf2, out);
}